// PGCN_64707977281948
// MI455X (gfx1250) — compile-verified
//
#include <hip/hip_runtime.h>
#include <hip/hip_bf16.h>

typedef __attribute__((ext_vector_type(2))) float v2f;
typedef __attribute__((ext_vector_type(8))) float v8f;
typedef unsigned int u32;
typedef __attribute__((ext_vector_type(4))) u32 u32x4;
typedef __attribute__((ext_vector_type(8))) u32 u32x8;

#define HD   128   // feature / hidden dim
#define MLPD 256   // MLP hidden dim
#define ASTR 132   // padded LDS row stride (floats): 16B-aligned, bank-conflict-free

// hardware f32 atomic add (global_atomic_add_f32, no CAS loop)
__device__ __forceinline__ void atomAddF(float* p, float v) {
  unsafeAtomicAdd(p, v);
}

// ---------------------------------------------------------------- utilities
__global__ __launch_bounds__(256) void k_zero(float* __restrict__ p, size_t n) {
  size_t i = (size_t)blockIdx.x * blockDim.x + threadIdx.x;
  if (i < n) p[i] = 0.0f;
}

__global__ __launch_bounds__(256) void k_scale_copy(float* __restrict__ dst,
                                                    const float* __restrict__ src,
                                                    float a, size_t n) {
  size_t i = (size_t)blockIdx.x * blockDim.x + threadIdx.x;
  if (i < n) dst[i] = a * src[i];
}

// ---------------------------------------------------------------- edge norm
__global__ __launch_bounds__(256) void k_deg(const int* __restrict__ src,
                                             const int* __restrict__ dst,
                                             const float* __restrict__ w,
                                             float* __restrict__ deg, int E) {
  int e = blockIdx.x * blockDim.x + threadIdx.x;
  if (e >= E) return;
  int s = src[e], d = dst[e];
  float wv = (s == d) ? 0.0f : w[e];
  if (wv != 0.0f) atomAddF(&deg[s], wv);
}

__global__ __launch_bounds__(256) void k_norm(const int* __restrict__ src,
                                              const int* __restrict__ dst,
                                              const float* __restrict__ w,
                                              const float* __restrict__ deg,
                                              float* __restrict__ norm, int E) {
  int e = blockIdx.x * blockDim.x + threadIdx.x;
  if (e >= E) return;
  int s = src[e], d = dst[e];
  float wv = (s == d) ? 0.0f : w[e];
  float ds_ = deg[s], dd = deg[d];
  float is = (ds_ > 0.0f) ? rsqrtf(ds_) : 0.0f;
  float id = (dd > 0.0f) ? rsqrtf(dd) : 0.0f;
  norm[e] = -is * wv * id;
}

// ------------------------------------------------- propagation (scatter-add)
// out[dst] += scale * norm[e] * x[src]; one wave32 per edge, float4 per lane.
__global__ __launch_bounds__(256) void k_scatter(const float* __restrict__ x,
                                                 const int* __restrict__ src,
                                                 const int* __restrict__ dst,
                                                 const float* __restrict__ norm,
                                                 float* __restrict__ out,
                                                 float scale, int E) {
  int gid  = blockIdx.x * blockDim.x + threadIdx.x;
  int e    = gid >> 5;          // wave-uniform
  int lane = gid & 31;
  if (e >= E) return;
  float nv = norm[e] * scale;
  if (nv == 0.0f) return;       // self loops / zero-weight edges contribute 0
  int s = src[e], d = dst[e];
  const float4 v = reinterpret_cast<const float4*>(x + (size_t)s * HD)[lane];
  float* o = out + (size_t)d * HD + lane * 4;
  atomAddF(o + 0, nv * v.x);
  atomAddF(o + 1, nv * v.y);
  atomAddF(o + 2, nv * v.z);
  atomAddF(o + 3, nv * v.w);
}

// ------------------------------------------------- TDM tile load to LDS
// DMA a 16-row x 128-float tile (row stride = tensor_dim0_stride floats) into
// LDS at lds_off, inserting a 4-DWORD pad after every 128 DWORDs so the LDS
// row stride becomes ASTR=132 floats.  Issued once per wave; TENSORcnt tracks.
__device__ __forceinline__ void tdm_load_tile(const float* gsrc, u32 lds_off,
                                              u32 rows_avail) {
  unsigned long long ga = (unsigned long long)(uintptr_t)gsrc;
  u32x4 g0 = { 1u,                                   // count=1, load, user D#
               lds_off,                               // D#.lds_addr
               (u32)ga,                               // global_addr[31:0]
               (u32)((ga >> 32) & 0x01FFFFFFu) | (2u << 30) }; // addr[56:32]|type=2
  u32x8 g1 = { (2u << 16)            // data_size = 4B
               | (1u << 20)          // pad_enable
               | (6u << 22)          // pad_interval: 128 DWORDs
               | (3u << 25),         // pad_amount:   4 DWORDs
               (u32)(HD & 0xFFFFu) << 16,                    // tensor_dim0 lo16
               ((u32)(HD >> 16) & 0xFFFFu)
                 | ((rows_avail & 0xFFFFu) << 16),           // dim0 hi / dim1 lo
               ((rows_avail >> 16) & 0xFFFFu)
                 | ((u32)HD << 16),                          // dim1 hi / tile_dim0
               16u,                                          // tile_dim1=16, tile_dim2=0
               (u32)HD,                                      // tensor_dim0_stride lo32
               0u, 0u };                                     // stride hi / dim1_stride
  u32x4 g2 = { 1u, 0u, 0u, 0u };    // tensor_dim2=1 (unused, tile_dim2=0)
  u32x4 g3 = { 0u, 0u, 0u, 0u };
  asm volatile("tensor_load_to_lds %0, %1, %2, %3"
               :: "s"(g0), "s"(g1), "s"(g2), "s"(g3)
               : "memory");
}

// --------------------------------------------------- fused ChebConv GEMM
// out = relu(X0@W[0] + X1@W[1] + X2@W[2]); W is (3,128,128) contiguous, i.e.
// the vertical concat along K. One block = one 16-row M tile; 8 waves each
// own one 16-col N tile. A tiles DMA'd into LDS by the Tensor Data Mover.
__global__ __launch_bounds__(256) void k_cheb_gemm(const float* __restrict__ X0,
                                                   const float* __restrict__ X1,
                                                   const float* __restrict__ X2,
                                                   const float* __restrict__ W,
                                                   float* __restrict__ out,
                                                   int nodes) {
  __shared__ float As[3 * 16 * ASTR];
  const int m0  = blockIdx.x * 16;
  const int tid = threadIdx.x;
  const float* Xs[3] = {X0, X1, X2};

  if (tid < 32) {                       // wave 0 drives the TDM
    u32 rows = (u32)(nodes - m0);       // OOB rows read back as zero
    u32 base = (u32)(uintptr_t)&As[0];  // flat LDS addr: low 32 bits = offset
    #pragma unroll
    for (int s = 0; s < 3; ++s)
      tdm_load_tile(Xs[s] + (size_t)m0 * HD, base + (u32)(s * 16 * ASTR * 4), rows);
    __builtin_amdgcn_s_wait_tensorcnt(0);
  }
  __syncthreads();

  const int wave = tid >> 5, lane = tid & 31;
  const int half = lane >> 4, r = lane & 15;
  const int n0   = wave * 16;

  v8f c = {};
  #pragma unroll
  for (int seg = 0; seg < 3; ++seg) {
    const float* Wp   = W + (size_t)seg * HD * HD;
    const float* Arow = &As[(seg * 16 + r) * ASTR];
    #pragma unroll 4
    for (int k = 0; k < HD; k += 4) {
      int kk = k + 2 * half;
      v2f a; a.x = Arow[kk];                 a.y = Arow[kk + 1];
      v2f b; b.x = Wp[kk * HD + n0 + r];     b.y = Wp[(kk + 1) * HD + n0 + r];
      c = __builtin_amdgcn_wmma_f32_16x16x4_f32(false, a, false, b,
                                                (short)0, c, false, false);
    }
  }

  // C tile layout: VGPR j, lanes 0-15 -> M=j, lanes 16-31 -> M=8+j; N = lane&15
  #pragma unroll
  for (int j = 0; j < 8; ++j) {
    int m = m0 + j + 8 * half;
    if (m < nodes) {
      float v = c[j];
      out[(size_t)m * HD + n0 + r] = v > 0.0f ? v : 0.0f;
    }
  }
}

// --------------------------------------------------- fused MLP head
// h = relu(X@w1 + b1); h = (h-mean)*rsqrt(var+eps)*gamma + beta;
// logit = h@w2 + b2.  Block = 16 nodes; 8 waves * 2 N-tiles cover MLPD=256.
__global__ __launch_bounds__(256) void k_mlp(const float* __restrict__ X,
                                             const float* __restrict__ w1,
                                             const float* __restrict__ b1,
                                             const float* __restrict__ gamma,
                                             const float* __restrict__ beta,
                                             const float* __restrict__ mean,
                                             const float* __restrict__ var,
                                             const float* __restrict__ w2,
                                             const float* __restrict__ b2,
                                             float* __restrict__ out, int nodes) {
  __shared__ float As[16 * ASTR];
  __shared__ float lg[32];      // [16 rows][2 classes]
  const int m0  = blockIdx.x * 16;
  const int tid = threadIdx.x;

  #pragma unroll
  for (int i = 0; i < 2; ++i) {
    int flat = tid + i * 256;   // 0..511
    int row  = flat >> 5;
    int q    = flat & 31;
    int rg   = m0 + row; if (rg >= nodes) rg = nodes - 1;
    float4 v = reinterpret_cast<const float4*>(X + (size_t)rg * HD)[q];
    *reinterpret_cast<float4*>(&As[row * ASTR + q * 4]) = v;
  }
  if (tid < 32) lg[tid] = 0.0f;
  __syncthreads();

  const int wave = tid >> 5, lane = tid & 31;
  const int half = lane >> 4, r = lane & 15;
  float pc0[8] = {0}, pc1[8] = {0};

  #pragma unroll
  for (int t = 0; t < 2; ++t) {
    int n0 = (wave * 2 + t) * 16;
    v8f c = {};
    const float* Arow = &As[r * ASTR];
    #pragma unroll 4
    for (int k = 0; k < HD; k += 4) {
      int kk = k + 2 * half;
      v2f a; a.x = Arow[kk];                  a.y = Arow[kk + 1];
      v2f b; b.x = w1[kk * MLPD + n0 + r];    b.y = w1[(kk + 1) * MLPD + n0 + r];
      c = __builtin_amdgcn_wmma_f32_16x16x4_f32(false, a, false, b,
                                                (short)0, c, false, false);
    }
    int n = n0 + r;
    float bb = b1[n], mu = mean[n], s = gamma[n] * rsqrtf(var[n] + 1e-5f);
    float be = beta[n], w20 = w2[n * 2 + 0], w21 = w2[n * 2 + 1];
    #pragma unroll
    for (int j = 0; j < 8; ++j) {
      float h = c[j] + bb;
      h = h > 0.0f ? h : 0.0f;
      h = (h - mu) * s + be;
      pc0[j] += h * w20;
      pc1[j] += h * w21;
    }
  }

  #pragma unroll
  for (int j = 0; j < 8; ++j) {
    int m = j + 8 * half;
    atomicAdd(&lg[m * 2 + 0], pc0[j]);   // ds_add_f32
    atomicAdd(&lg[m * 2 + 1], pc1[j]);
  }
  __syncthreads();
  if (tid < 32) {
    int m = m0 + (tid >> 1);
    if (m < nodes) out[(size_t)m * 2 + (tid & 1)] = lg[tid] + b2[tid & 1];
  }
}

// ---------------------------------------------------------------- launcher
extern "C" void kernel_launch(void* const* d_in, const int* in_sizes, int n_in,
                              void* d_out, int out_size, void* d_ws, size_t ws_size,
                              hipStream_t stream) {
  const float* features = (const float*)d_in[0];
  const float* edge_w   = (const float*)d_in[1];
  const float* cw[3]    = {(const float*)d_in[2], (const float*)d_in[3],
                           (const float*)d_in[4]};
  const float* cls_w1 = (const float*)d_in[5];
  const float* cls_b1 = (const float*)d_in[6];
  const float* bn_g   = (const float*)d_in[7];
  const float* bn_b   = (const float*)d_in[8];
  const float* bn_m   = (const float*)d_in[9];
  const float* bn_v   = (const float*)d_in[10];
  const float* cls_w2 = (const float*)d_in[11];
  const float* cls_b2 = (const float*)d_in[12];
  const int*   eidx   = (const int*)d_in[13];

  const int E = in_sizes[1];
  const int N = in_sizes[0] / HD;
  const int* src = eidx;
  const int* dst = eidx + E;

  const size_t NF = (size_t)N * HD;
  float* deg  = (float*)d_ws;
  float* norm = deg + N;
  float* buf0 = norm + E;
  float* buf1 = buf0 + NF;
  float* buf2 = buf1 + NF;
  float* buf3 = buf2 + NF;

  const int THR = 256;
  dim3 blk(THR);
  int gE   = (E + THR - 1) / THR;
  int gE32 = (int)(((size_t)E * 32 + THR - 1) / THR);
  int gNF  = (int)((NF + THR - 1) / THR);
  int gN   = (N + THR - 1) / THR;
  int gMt  = (N + 15) / 16;

  // edge normalization
  k_zero<<<gN, blk, 0, stream>>>(deg, (size_t)N);
  k_deg<<<gE, blk, 0, stream>>>(src, dst, edge_w, deg, E);
  k_norm<<<gE, blk, 0, stream>>>(src, dst, edge_w, deg, norm, E);

  // three ChebConv layers (buffer rotation: in -> out)
  const float* xin = features;
  float* outs[3] = {buf0, buf3, buf0};
  for (int l = 0; l < 3; ++l) {
    float* O = outs[l];
    // Tx1 = prop(x)
    k_zero<<<gNF, blk, 0, stream>>>(buf1, NF);
    k_scatter<<<gE32, blk, 0, stream>>>(xin, src, dst, norm, buf1, 1.0f, E);
    // Tx2 = 2*prop(Tx1) - Tx0
    k_scale_copy<<<gNF, blk, 0, stream>>>(buf2, xin, -1.0f, NF);
    k_scatter<<<gE32, blk, 0, stream>>>(buf1, src, dst, norm, buf2, 2.0f, E);
    // out = relu(Tx0@W0 + Tx1@W1 + Tx2@W2)
    k_cheb_gemm<<<gMt, blk, 0, stream>>>(xin, buf1, buf2, cw[l], O, N);
    xin = O;
  }

  // fused MLP + BN + classifier -> logits at d_out[0 .. N*2)
  k_mlp<<<gMt, blk, 0, stream>>>(xin, cls_w1, cls_b1, bn_g, bn_b, bn_m, bn_v,
                                 cls_w2, cls_b2, (float*)d_out, N);

  // second tuple element: edge_weight passthrough at d_out[N*2 .. N*2+E)
  k_scale_copy<<<gE, blk, 0, stream>>>((float*)d_out + (size_t)N * 2,
                                       edge_w, 1.0f, (size_t)E);
}